// GCGRUCell_29764123361454
// MI455X (gfx1250) — compile-verified
//
#include <hip/hip_runtime.h>
#include <hip/hip_bf16.h>

#define N_NODES 50000
#define N_EDGES 800000
#define DIM     128          // IN_DIM == HIDDEN == 128
#define COLS    384          // 3 * HIDDEN
#define LDSPAD  132          // padded row stride (floats) to avoid LDS bank conflicts
#define MTILES  2            // 16-node M tiles per block (32 nodes/block)

typedef __attribute__((ext_vector_type(2))) float v2f;
typedef __attribute__((ext_vector_type(8))) float v8f;

__device__ __forceinline__ float fast_sigmoid(float x) {
    // 1 / (1 + e^-x) with hardware v_rcp_f32
    return __builtin_amdgcn_rcpf(1.0f + __expf(-x));
}
__device__ __forceinline__ float fast_tanh(float x) {
    // tanh(x) = 1 - 2/(e^{2x}+1)
    return 1.0f - 2.0f * __builtin_amdgcn_rcpf(__expf(2.0f * x) + 1.0f);
}

// ---------------------------------------------------------------- zero scratch
__global__ void gcgru_zero(float* ws, long n) {
    long i = (long)blockIdx.x * blockDim.x + threadIdx.x;
    long stride = (long)gridDim.x * blockDim.x;
    for (; i < n; i += stride) ws[i] = 0.0f;
}

// ---------------------------------------------------------------- degree count
__global__ void gcgru_degrees(const int* __restrict__ src, const int* __restrict__ dst,
                              float* deg_out, float* deg_in, int n_edges) {
    int e = blockIdx.x * blockDim.x + threadIdx.x;
    if (e < n_edges) {
        atomicAdd(&deg_out[src[e]], 1.0f);
        atomicAdd(&deg_in [dst[e]], 1.0f);
    }
}

// ---------------------------------------------------------------- deg -> rsqrt norm (in place)
__global__ void gcgru_norms(float* deg_out, float* deg_in, int n_nodes) {
    int i = blockIdx.x * blockDim.x + threadIdx.x;
    if (i < n_nodes) {
        float a = deg_out[i]; deg_out[i] = rsqrtf(a < 1.0f ? 1.0f : a);
        float b = deg_in[i];  deg_in[i]  = rsqrtf(b < 1.0f ? 1.0f : b);
    }
}

// ---------------------------------------------------------------- edge scatter-add
// 32 threads per edge, each handles 4 contiguous dims (float4 gather, 4 f32 atomics)
// for both feat->agg_f and hx->agg_h.
__global__ void gcgru_scatter(const float* __restrict__ feat, const float* __restrict__ hx,
                              const int* __restrict__ src, const int* __restrict__ dst,
                              const float* __restrict__ norm_src,
                              float* agg_f, float* agg_h, int n_edges) {
    long gid = (long)blockIdx.x * blockDim.x + threadIdx.x;
    int e   = (int)(gid >> 5);
    int sub = (int)(gid & 31);        // dim group: 4 floats each
    if (e >= n_edges) return;
    int s = src[e];
    int d = dst[e];
    float ns = norm_src[s];
    const float4* f4 = (const float4*)feat;
    const float4* h4 = (const float4*)hx;
    float4 vf = f4[(long)s * (DIM / 4) + sub];
    float4 vh = h4[(long)s * (DIM / 4) + sub];
    long ob = (long)d * DIM + sub * 4;
    atomicAdd(&agg_f[ob + 0], vf.x * ns);
    atomicAdd(&agg_f[ob + 1], vf.y * ns);
    atomicAdd(&agg_f[ob + 2], vf.z * ns);
    atomicAdd(&agg_f[ob + 3], vf.w * ns);
    atomicAdd(&agg_h[ob + 0], vh.x * ns);
    atomicAdd(&agg_h[ob + 1], vh.y * ns);
    atomicAdd(&agg_h[ob + 2], vh.z * ns);
    atomicAdd(&agg_h[ob + 3], vh.w * ns);
}

// ---------------------------------------------------------------- fused GEMM + GRU gate
// Block = 256 threads (8 waves), one block per 32-node tile (2 x 16-row M tiles).
// LDS holds the 32x128 agg tiles (scaled by norm_dst) for feat and hx paths.
// Wave w owns output column group j = w (16 of the 128 hidden dims). Per K step
// it loads one B fragment per {W_i,W_h}x{r,z,n} and reuses it for both M tiles:
// 12 x V_WMMA_F32_16X16X4_F32 per K step, 1 global B load per WMMA.
__global__ __launch_bounds__(256)
void gcgru_gemm(const float* __restrict__ hx,
                const float* __restrict__ W_i, const float* __restrict__ b_i,
                const float* __restrict__ W_h, const float* __restrict__ b_h,
                const float* __restrict__ agg_f, const float* __restrict__ agg_h,
                const float* __restrict__ norm_dst,
                float* __restrict__ out) {
    __shared__ float sf[16 * MTILES * LDSPAD];
    __shared__ float sh[16 * MTILES * LDSPAD];

    const int tile = blockIdx.x;            // 32-node tile
    const int row0 = tile * 16 * MTILES;    // first node of block

    // ---- stage agg tiles (x norm_dst) into LDS
    for (int i = threadIdx.x; i < 16 * MTILES * DIM; i += 256) {
        int m = i >> 7;          // node row in block (0..31)
        int k = i & (DIM - 1);   // feature dim
        int node = row0 + m;
        float vf = 0.0f, vh = 0.0f;
        if (node < N_NODES) {
            float nd = norm_dst[node];
            vf = agg_f[(long)node * DIM + k] * nd;
            vh = agg_h[(long)node * DIM + k] * nd;
        }
        sf[m * LDSPAD + k] = vf;
        sh[m * LDSPAD + k] = vh;
    }
    __syncthreads();

    const int lane  = threadIdx.x & 31;
    const int j     = threadIdx.x >> 5;     // wave id == column group 0..7
    const int nlo   = lane & 15;            // column (and A-row) within 16
    const int khalf = (lane >> 4) << 1;     // K sub-offset: 0 or 2

    // ---- bias-initialized accumulators (bias depends only on column)
    v8f acc_i[3][MTILES], acc_h[3][MTILES];
#pragma unroll
    for (int g = 0; g < 3; ++g) {
        float bi = b_i[g * DIM + j * 16 + nlo];
        float bh = b_h[g * DIM + j * 16 + nlo];
#pragma unroll
        for (int t = 0; t < MTILES; ++t)
#pragma unroll
            for (int v = 0; v < 8; ++v) { acc_i[g][t][v] = bi; acc_h[g][t][v] = bh; }
    }

    // ---- K loop: 12 WMMAs per step, B fragments shared across both M tiles
    for (int k0 = 0; k0 < DIM; k0 += 4) {
        int kk = k0 + khalf;
        v2f af[MTILES], ah[MTILES];
#pragma unroll
        for (int t = 0; t < MTILES; ++t) {
            int row = (nlo + 16 * t) * LDSPAD + kk;
            af[t].x = sf[row];     af[t].y = sf[row + 1];
            ah[t].x = sh[row];     ah[t].y = sh[row + 1];
        }
#pragma unroll
        for (int g = 0; g < 3; ++g) {
            int n = g * DIM + j * 16 + nlo;
            v2f bi, bh2;
            bi.x  = W_i[(long)kk * COLS + n];
            bi.y  = W_i[(long)(kk + 1) * COLS + n];
            bh2.x = W_h[(long)kk * COLS + n];
            bh2.y = W_h[(long)(kk + 1) * COLS + n];
#pragma unroll
            for (int t = 0; t < MTILES; ++t) {
                acc_i[g][t] = __builtin_amdgcn_wmma_f32_16x16x4_f32(
                    false, af[t], false, bi,  (short)0, acc_i[g][t], false, false);
                acc_h[g][t] = __builtin_amdgcn_wmma_f32_16x16x4_f32(
                    false, ah[t], false, bh2, (short)0, acc_h[g][t], false, false);
            }
        }
    }

    // ---- GRU gating + store (fast rcp-based activations)
    const int mbase = (lane >> 4) << 3;     // 0 or 8
#pragma unroll
    for (int t = 0; t < MTILES; ++t) {
#pragma unroll
        for (int v = 0; v < 8; ++v) {
            int node = row0 + 16 * t + mbase + v;
            float r  = fast_sigmoid(acc_i[0][t][v] + acc_h[0][t][v]);
            float z  = fast_sigmoid(acc_i[1][t][v] + acc_h[1][t][v]);
            float nn = fast_tanh(acc_i[2][t][v] + r * acc_h[2][t][v]);
            if (node < N_NODES) {
                long idx = (long)node * DIM + j * 16 + nlo;
                float hxv = hx[idx];
                out[idx] = (1.0f - z) * nn + z * hxv;
            }
        }
    }
}

// ---------------------------------------------------------------- launch
extern "C" void kernel_launch(void* const* d_in, const int* in_sizes, int n_in,
                              void* d_out, int out_size, void* d_ws, size_t ws_size,
                              hipStream_t stream) {
    const float* feat = (const float*)d_in[0];
    const float* hx   = (const float*)d_in[1];
    const float* W_i  = (const float*)d_in[2];
    const float* b_i  = (const float*)d_in[3];
    const float* W_h  = (const float*)d_in[4];
    const float* b_h  = (const float*)d_in[5];
    const int*   src  = (const int*)d_in[6];
    const int*   dst  = (const int*)d_in[7];
    float*       out  = (float*)d_out;

    // workspace layout (floats): norm_src | norm_dst | agg_f | agg_h
    float* ws       = (float*)d_ws;
    float* norm_src = ws;
    float* norm_dst = ws + N_NODES;
    float* agg_f    = ws + 2 * N_NODES;
    float* agg_h    = agg_f + (long)N_NODES * DIM;

    const long n_zero = 2L * N_NODES + 2L * N_NODES * DIM;
    gcgru_zero<<<4096, 256, 0, stream>>>(ws, n_zero);

    gcgru_degrees<<<(N_EDGES + 255) / 256, 256, 0, stream>>>(src, dst, norm_src, norm_dst, N_EDGES);
    gcgru_norms<<<(N_NODES + 255) / 256, 256, 0, stream>>>(norm_src, norm_dst, N_NODES);

    long sc_threads = (long)N_EDGES * 32;
    int  sc_blocks  = (int)((sc_threads + 255) / 256);
    gcgru_scatter<<<sc_blocks, 256, 0, stream>>>(feat, hx, src, dst, norm_src,
                                                 agg_f, agg_h, N_EDGES);

    int tiles = (N_NODES + 16 * MTILES - 1) / (16 * MTILES);
    gcgru_gemm<<<tiles, 256, 0, stream>>>(hx, W_i, b_i, W_h, b_h,
                                          agg_f, agg_h, norm_dst, out);
}